// RLFractalFrameGenerator_65395172049476
// MI455X (gfx1250) — compile-verified
//
#include <hip/hip_runtime.h>

// Problem constants (match reference).
constexpr int NUM_FRAMES = 64;
constexpr int H = 2048;
constexpr int W = 2048;
constexpr int MAX_DEPTH = 12;

constexpr int BLOCK = 256;          // 8 waves on wave32 CDNA5
constexpr int PX_PER_THREAD = 4;    // -> b128 stores per plane
constexpr int TAB_ENTRIES = 2 * NUM_FRAMES;  // 64 node int4 + 64 color float4, contiguous

typedef __attribute__((ext_vector_type(4))) float v4f;

#if defined(__gfx1250__) && __has_builtin(__builtin_amdgcn_global_load_async_to_lds_b128)
#define HAVE_ASYNC_LDS 1
#else
#define HAVE_ASYNC_LDS 0
#endif

#if HAVE_ASYNC_LDS
// Exact pointee type the builtin expects (per hipcc diagnostic):
//   __attribute__((__vector_size__(16))) int
typedef int b128_t __attribute__((__vector_size__(16)));
typedef __attribute__((address_space(1))) b128_t* glb_b128_ptr;
typedef __attribute__((address_space(3))) b128_t* lds_b128_ptr;
#endif

// ---------------------------------------------------------------------------
// Phase 1: build the per-frame descent table into one contiguous 2KB region:
//   tab[f]              = {left_child, right_child, ratio_bits, 0}   (int4)
//   tab[NUM_FRAMES + f] = {r, g, b, 0}                               (float4 bits)
// Tiny: 64 lanes, each scans 2x64 logits (first-max tie-break == jnp.argmax).
// ---------------------------------------------------------------------------
__global__ __launch_bounds__(64) void prep_kernel(
    const float* __restrict__ frame_colors,     // [64,3]
    const float* __restrict__ frame_selection,  // [64,2,64]
    const float* __restrict__ split_ratios,     // [64]
    int4* __restrict__ tab)                     // ws: [128] x 16B
{
    const int f = threadIdx.x;
    if (f >= NUM_FRAMES) return;

    const float* s0 = frame_selection + (size_t)f * 2 * NUM_FRAMES;
    const float* s1 = s0 + NUM_FRAMES;

    int b0 = 0, b1 = 0;
    float m0 = s0[0], m1 = s1[0];
    #pragma unroll 4
    for (int j = 1; j < NUM_FRAMES; ++j) {
        float v0 = s0[j];
        if (v0 > m0) { m0 = v0; b0 = j; }   // strict '>' keeps first max
        float v1 = s1[j];
        if (v1 > m1) { m1 = v1; b1 = j; }
    }

    tab[f] = make_int4(b0, b1, __float_as_int(split_ratios[f]), 0);
    tab[NUM_FRAMES + f] = make_int4(__float_as_int(frame_colors[3 * f + 0]),
                                    __float_as_int(frame_colors[3 * f + 1]),
                                    __float_as_int(frame_colors[3 * f + 2]), 0);
}

// ---------------------------------------------------------------------------
// Phase 2: per-pixel BSP descent. Each thread owns 4 consecutive x pixels.
// Table staged into LDS via GLOBAL_LOAD_ASYNC_TO_LDS_B128 (ASYNCcnt path)
// when the builtin exists; ds_load_b128 gathers in the loop; NT b128 stores.
// ---------------------------------------------------------------------------
__global__ __launch_bounds__(BLOCK) void render_kernel(
    const int4* __restrict__ tab,               // [128] x 16B in ws
    float* __restrict__ out)                    // [3,H,W]
{
    __shared__ int4 sTab[TAB_ENTRIES];          // [0,64): nodes, [64,128): colors

#if HAVE_ASYNC_LDS
    // CDNA5 async global->LDS: no VGPR staging, tracked by ASYNCcnt.
    // Builtin signature (from diagnostic): (global src, lds dst, imm off, imm cpol)
    if (threadIdx.x < TAB_ENTRIES) {
        glb_b128_ptr gsrc = (glb_b128_ptr)(tab + threadIdx.x);
        lds_b128_ptr ldst = (lds_b128_ptr)(&sTab[threadIdx.x]);
        __builtin_amdgcn_global_load_async_to_lds_b128(gsrc, ldst, 0, 0);
#if __has_builtin(__builtin_amdgcn_s_wait_asynccnt)
        __builtin_amdgcn_s_wait_asynccnt(0);
#else
        asm volatile("s_wait_asynccnt 0x0" ::: "memory");
#endif
    }
    __syncthreads();
#else
    if (threadIdx.x < TAB_ENTRIES) {
        sTab[threadIdx.x] = tab[threadIdx.x];
    }
    __syncthreads();
#endif

    const int4*   sNode = sTab;
    const float4* sCol  = (const float4*)(sTab + NUM_FRAMES);

    const int y     = blockIdx.y;
    const int xbase = (blockIdx.x * BLOCK + threadIdx.x) * PX_PER_THREAD;

    int  idx[PX_PER_THREAD]  = {0, 0, 0, 0};
    int  x0[PX_PER_THREAD]   = {0, 0, 0, 0};
    int  y0[PX_PER_THREAD]   = {0, 0, 0, 0};
    int  w[PX_PER_THREAD]    = {W, W, W, W};
    int  h[PX_PER_THREAD]    = {H, H, H, H};
    bool done[PX_PER_THREAD] = {false, false, false, false};

    for (int d = 0; d < MAX_DEPTH; ++d) {
        #pragma unroll
        for (int p = 0; p < PX_PER_THREAD; ++p) {
            // Leaf check BEFORE update (matches reference semantics).
            const bool leaf = (w[p] < 2) | (h[p] < 2);
            const bool act  = (!done[p]) & (!leaf);
            done[p] = done[p] | leaf;

            const int4  nd    = sNode[idx[p]];
            const float ratio = __int_as_float(nd.z);
            const bool  isv   = (idx[p] & 1) != 0;   // odd frame -> vertical split

            // Only the active axis matters this level.
            const int size = isv ? w[p]  : h[p];
            const int org  = isv ? x0[p] : y0[p];
            const int pos  = isv ? (xbase + p) : y;

            int cut = (int)((float)size * ratio);    // trunc == floor (size>=0)
            cut = (cut < 1) ? 1 : cut;

            const bool goleft = (pos - org) < cut;
            const int  child  = goleft ? nd.x : nd.y;
            const int  norg   = goleft ? org  : (org + cut);
            const int  nsz    = goleft ? cut  : (size - cut);

            if (act) {
                idx[p] = child;
                if (isv) { x0[p] = norg; w[p] = nsz; }
                else     { y0[p] = norg; h[p] = nsz; }
            }
        }
        // Wave-uniform early exit: consecutive pixels share tree paths, so
        // whole waves hit all-leaf states well before depth 12.
        const int alive = (!done[0]) | (!done[1]) | (!done[2]) | (!done[3]);
        if (!__any(alive)) break;
    }

    // Gather colors from LDS and transpose into per-plane float4 runs.
    const float4 c0 = sCol[idx[0]];
    const float4 c1 = sCol[idx[1]];
    const float4 c2 = sCol[idx[2]];
    const float4 c3 = sCol[idx[3]];

    const long long plane = (long long)H * W;
    float* base = out + (long long)y * W + xbase;   // 16B aligned (xbase % 4 == 0)

    const v4f r = {c0.x, c1.x, c2.x, c3.x};
    const v4f g = {c0.y, c1.y, c2.y, c3.y};
    const v4f b = {c0.z, c1.z, c2.z, c3.z};

    // Write-once 48 MiB stream: non-temporal hint keeps near caches clean.
    __builtin_nontemporal_store(r, (v4f*)(base));
    __builtin_nontemporal_store(g, (v4f*)(base + plane));
    __builtin_nontemporal_store(b, (v4f*)(base + 2 * plane));
}

// ---------------------------------------------------------------------------
extern "C" void kernel_launch(void* const* d_in, const int* in_sizes, int n_in,
                              void* d_out, int out_size, void* d_ws, size_t ws_size,
                              hipStream_t stream) {
    (void)in_sizes; (void)n_in; (void)out_size; (void)ws_size;

    const float* frame_colors    = (const float*)d_in[0]; // [64,3]
    const float* frame_selection = (const float*)d_in[1]; // [64,2,64]
    const float* split_ratios    = (const float*)d_in[2]; // [64]
    float*       out             = (float*)d_out;         // [3,2048,2048]

    int4* tab = (int4*)d_ws;                              // 128 x 16B

    prep_kernel<<<1, 64, 0, stream>>>(frame_colors, frame_selection,
                                      split_ratios, tab);

    dim3 grid(W / (BLOCK * PX_PER_THREAD), H);  // (2, 2048)
    render_kernel<<<grid, BLOCK, 0, stream>>>(tab, out);
}